// GCNModel_82325933130193
// MI455X (gfx1250) — compile-verified
//
#include <hip/hip_runtime.h>
#include <hip/hip_bf16.h>

typedef float v2f __attribute__((ext_vector_type(2)));
typedef float v8f __attribute__((ext_vector_type(8)));

#define FDIM 128          // feature dim (F == H == 128)
#define CDIM 32           // classifier outputs
#define KHALF 64          // W staged into LDS in two 64-row halves
#define WSTRIDE 136       // LDS row stride (words): 2*136 mod 64 == 16 -> lane halves hit disjoint banks

// ---------------------------------------------------------------------------
// Degree / normalization kernels
// ---------------------------------------------------------------------------
__global__ void set_ones_kernel(float* __restrict__ deg, int n) {
    int i = blockIdx.x * blockDim.x + threadIdx.x;
    if (i < n) deg[i] = 1.0f;               // self-loop contributes 1 to every degree
}

__global__ void deg_accum_kernel(const int* __restrict__ edge, float* __restrict__ deg, int E) {
    int e = blockIdx.x * blockDim.x + threadIdx.x;
    if (e < E) atomicAdd(&deg[edge[(size_t)E + e]], 1.0f);   // count dst occurrences
}

__global__ void deg_to_dinv_kernel(float* __restrict__ deg, int n) {
    int i = blockIdx.x * blockDim.x + threadIdx.x;
    if (i < n) deg[i] = rsqrtf(deg[i]);     // deg >= 1 always (self loop)
}

// ---------------------------------------------------------------------------
// FP32 WMMA GEMM:  O[M x 128] = A[M x 128] @ W[128 x 128]
// Block: 256 threads (8 wave32s). Block -> one 16-row M tile, wave w -> cols 16w..16w+15.
// Uses V_WMMA_F32_16X16X4_F32; W staged in LDS in two 64-row halves.
// ---------------------------------------------------------------------------
__global__ void gemm_wmma_f32_kernel(const float* __restrict__ A,
                                     const float* __restrict__ W,
                                     float* __restrict__ O, int M) {
    __shared__ float Wlds[KHALF * WSTRIDE];

    const int lane  = threadIdx.x & 31;
    const int wave  = threadIdx.x >> 5;     // 0..7 -> N tile
    const int hl    = lane >> 4;            // lane half: 0 or 1
    const int l15   = lane & 15;
    const int row0  = blockIdx.x * 16;
    const int row   = min(row0 + l15, M - 1);   // clamp (keeps EXEC all-ones for WMMA)
    const int col   = wave * 16 + l15;
    const int klo   = hl * 2;               // lanes 0-15: K=k,k+1 ; lanes 16-31: K=k+2,k+3

    v8f c = {};
    for (int kh = 0; kh < 2; ++kh) {
        __syncthreads();
        // cooperatively stage W[kh*64 .. kh*64+63][0..127] into LDS
        for (int idx = threadIdx.x; idx < KHALF * FDIM; idx += blockDim.x) {
            int kr = idx >> 7;              // 0..63
            int kc = idx & (FDIM - 1);      // 0..127
            Wlds[kr * WSTRIDE + kc] = W[(size_t)(kh * KHALF + kr) * FDIM + kc];
        }
        __syncthreads();

        const int kbase = kh * KHALF;
        for (int k0 = 0; k0 < KHALF; k0 += 4) {
            // A operand: 16x4 fp32, row-major per ISA layout
            const float2 av = *(const float2*)(A + (size_t)row * FDIM + kbase + k0 + klo);
            v2f a; a.x = av.x; a.y = av.y;
            // B operand: 4x16 fp32 from LDS
            v2f b;
            b.x = Wlds[(k0 + klo)     * WSTRIDE + col];
            b.y = Wlds[(k0 + klo + 1) * WSTRIDE + col];
            c = __builtin_amdgcn_wmma_f32_16x16x4_f32(false, a, false, b,
                                                      (short)0, c, false, false);
        }
    }

    // C/D layout: VGPR g -> row row0+g (lanes 0-15) / row0+g+8 (lanes 16-31), N = col
    for (int g = 0; g < 8; ++g) {
        int r = row0 + g + hl * 8;
        if (r < M) O[(size_t)r * FDIM + col] = c[g];
    }
}

// ---------------------------------------------------------------------------
// Aggregation kernels
// ---------------------------------------------------------------------------
// agg[i,:] = h[i,:] * dinv[i]^2   (self-loop term; also initializes agg)
__global__ void selfloop_kernel(const float* __restrict__ H, const float* __restrict__ dinv,
                                float* __restrict__ Agg, int n) {
    const int t = threadIdx.x;              // 128 threads = feature dim
    for (int i = blockIdx.x; i < n; i += gridDim.x) {
        float d = dinv[i];
        Agg[(size_t)i * FDIM + t] = H[(size_t)i * FDIM + t] * d * d;
    }
}

// per-edge: agg[dst,:] += h[src,:] * dinv[src]*dinv[dst]
__global__ void edge_scatter_kernel(const int* __restrict__ edge,
                                    const float* __restrict__ H,
                                    const float* __restrict__ dinv,
                                    float* __restrict__ Agg, int E) {
    const int t = threadIdx.x;              // 128 threads = feature dim
    for (int e = blockIdx.x; e < E; e += gridDim.x) {
        int s = edge[e];
        int d = edge[(size_t)E + e];
        float norm = dinv[s] * dinv[d];
        atomicAdd(&Agg[(size_t)d * FDIM + t], H[(size_t)s * FDIM + t] * norm);
    }
}

__global__ void bias_relu_kernel(float* __restrict__ X, const float* __restrict__ b, int n) {
    const int t = threadIdx.x;
    const float bt = b[t];
    for (int i = blockIdx.x; i < n; i += gridDim.x) {
        float v = X[(size_t)i * FDIM + t] + bt;
        X[(size_t)i * FDIM + t] = v > 0.0f ? v : 0.0f;
    }
}

// ---------------------------------------------------------------------------
// Pooling + classifier
// ---------------------------------------------------------------------------
__global__ void zero_vec_kernel(float* __restrict__ p, int n) {
    int i = blockIdx.x * blockDim.x + threadIdx.x;
    if (i < n) p[i] = 0.0f;
}

__global__ void colsum_kernel(const float* __restrict__ X, float* __restrict__ pooled, int n) {
    const int t = threadIdx.x;              // 128
    float acc = 0.0f;
    for (int i = blockIdx.x; i < n; i += gridDim.x)
        acc += X[(size_t)i * FDIM + t];
    atomicAdd(&pooled[t], acc);
}

// out[c] = (sum_j pooledSum[j] * Wc[j,c]) / N + bc[c]
__global__ void classifier_kernel(const float* __restrict__ pooledSum,
                                  const float* __restrict__ Wc,
                                  const float* __restrict__ bc,
                                  float* __restrict__ out, float invN) {
    int c = threadIdx.x;                    // 32
    float acc = 0.0f;
    for (int j = 0; j < FDIM; ++j)
        acc += pooledSum[j] * Wc[(size_t)j * CDIM + c];
    out[c] = acc * invN + bc[c];
}

// ---------------------------------------------------------------------------
// Launch
// ---------------------------------------------------------------------------
extern "C" void kernel_launch(void* const* d_in, const int* in_sizes, int n_in,
                              void* d_out, int out_size, void* d_ws, size_t ws_size,
                              hipStream_t stream) {
    const float* x    = (const float*)d_in[0];
    const int*   edge = (const int*)  d_in[1];   // [2, E] int32 (JAX default x64-off)
    const float* W1   = (const float*)d_in[2];
    const float* b1   = (const float*)d_in[3];
    const float* W2   = (const float*)d_in[4];
    const float* b2   = (const float*)d_in[5];
    const float* Wc   = (const float*)d_in[6];
    const float* bc   = (const float*)d_in[7];
    float*       out  = (float*)d_out;

    const int N = in_sizes[0] / FDIM;            // 100000
    const int E = in_sizes[1] / 2;               // 1600000

    // workspace layout (floats): bufA | bufB | dinv | pooled
    float* bufA   = (float*)d_ws;
    float* bufB   = bufA + (size_t)N * FDIM;
    float* dinv   = bufB + (size_t)N * FDIM;
    float* pooled = dinv + N;

    const int mtiles = (N + 15) / 16;

    // --- degrees -> dinv ---
    set_ones_kernel   <<<(N + 255) / 256, 256, 0, stream>>>(dinv, N);
    deg_accum_kernel  <<<(E + 255) / 256, 256, 0, stream>>>(edge, dinv, E);
    deg_to_dinv_kernel<<<(N + 255) / 256, 256, 0, stream>>>(dinv, N);

    // --- layer 1: h1 = relu(Dinv A Dinv (x @ W1) + b1) ---
    gemm_wmma_f32_kernel<<<mtiles, 256, 0, stream>>>(x, W1, bufA, N);
    selfloop_kernel     <<<2048, FDIM, 0, stream>>>(bufA, dinv, bufB, N);
    edge_scatter_kernel <<<8192, FDIM, 0, stream>>>(edge, bufA, dinv, bufB, E);
    bias_relu_kernel    <<<2048, FDIM, 0, stream>>>(bufB, b1, N);

    // --- layer 2 ---
    gemm_wmma_f32_kernel<<<mtiles, 256, 0, stream>>>(bufB, W2, bufA, N);
    selfloop_kernel     <<<2048, FDIM, 0, stream>>>(bufA, dinv, bufB, N);
    edge_scatter_kernel <<<8192, FDIM, 0, stream>>>(edge, bufA, dinv, bufB, E);
    bias_relu_kernel    <<<2048, FDIM, 0, stream>>>(bufB, b2, N);

    // --- mean pool + classifier ---
    zero_vec_kernel   <<<1, FDIM, 0, stream>>>(pooled, FDIM);
    colsum_kernel     <<<1024, FDIM, 0, stream>>>(bufB, pooled, N);
    classifier_kernel <<<1, CDIM, 0, stream>>>(pooled, Wc, bc, out, 1.0f / (float)N);
}